// MinNormSolverFW_68624987456031
// MI455X (gfx1250) — compile-verified
//
#include <hip/hip_runtime.h>
#include <hip/hip_bf16.h>

typedef float v2f __attribute__((ext_vector_type(2)));
typedef float v8f __attribute__((ext_vector_type(8)));

#define FW_NTASKS   8
#define FW_MAXITER  250
#define FW_STOP     1e-6f

// ---------------------------------------------------------------------------
// Kernel 1: streaming Gram matrix via V_WMMA_F32_16X16X4_F32 (wave32).
//
// A (16x4 f32) register layout: lane L, vgpr r holds A(M = L&15, K = 2*(L>>4)+r)
// B (4x16 f32) register layout: lane L, vgpr r holds B(K = 2*(L>>4)+r, N = L&15)
// For B = A^T (Gram), both operands have identical per-lane contents, so one
// b64 load per lane feeds A and B.
//
// Dual-chunk packing: rows/cols 0-7 of the 16x16 tile hold tasks 0-7 at
// columns [k, k+4); rows/cols 8-15 hold the SAME tasks at columns [k+4, k+8).
// G[i][j] = C(i,j) + C(i+8, j+8); off-diagonal 8x8 blocks are discarded.
// ---------------------------------------------------------------------------
__global__ __launch_bounds__(256)
void gram_wmma_kernel(const float* __restrict__ vecs,
                      float* __restrict__ partials,
                      long D, int nwg) {
    __shared__ float lds_g[64];
    const int tid  = threadIdx.x;
    const int lane = tid & 31;
    // Wave index is wave-uniform: pin it to an SGPR so the k-loop gets scalar
    // control flow (s_add/s_cmp) instead of a vector counter + exec masking.
    const int wave = __builtin_amdgcn_readfirstlane(tid >> 5);   // 0..7
    const long wid = (long)blockIdx.x * 8 + wave;                // global wave id
    const long total_waves = (long)nwg * 8;

    if (tid < 64) lds_g[tid] = 0.0f;
    __syncthreads();

    // Per-lane fragment address for this wave's WMMA tiles.
    const int r16   = lane & 15;         // row of A / col of B in the 16x16 tile
    const int task  = r16 & 7;           // real task index (0..7)
    const int chunk = (r16 >> 3) & 1;    // 0: cols [k,k+4), 1: cols [k+4,k+8)
    const int h     = lane >> 4;         // K-half select
    const long coloff = (long)(chunk * 4 + h * 2);
    const float* rowp = vecs + (size_t)task * (size_t)D + coloff;

    v8f c = {};                          // 16x16 f32 accumulator (8 VGPRs)

    // Each iteration consumes 64 columns (8 WMMAs x 8 cols each via chunking):
    // 8 outstanding b64 loads per wave per iteration for memory-level
    // parallelism; all share one address VGPR pair via immediate offsets.
    const long stride = total_waves * 64;
    for (long k = wid * 64; k + 64 <= D; k += stride) {
        const float* p = rowp + k;
        __builtin_prefetch(p + stride, 0, 0);        // global_prefetch_b8
        v2f a0 = *(const v2f*)(p);
        v2f a1 = *(const v2f*)(p + 8);
        v2f a2 = *(const v2f*)(p + 16);
        v2f a3 = *(const v2f*)(p + 24);
        v2f a4 = *(const v2f*)(p + 32);
        v2f a5 = *(const v2f*)(p + 40);
        v2f a6 = *(const v2f*)(p + 48);
        v2f a7 = *(const v2f*)(p + 56);
        c = __builtin_amdgcn_wmma_f32_16x16x4_f32(false, a0, false, a0,
                                                  (short)0, c, false, false);
        c = __builtin_amdgcn_wmma_f32_16x16x4_f32(false, a1, false, a1,
                                                  (short)0, c, false, false);
        c = __builtin_amdgcn_wmma_f32_16x16x4_f32(false, a2, false, a2,
                                                  (short)0, c, false, false);
        c = __builtin_amdgcn_wmma_f32_16x16x4_f32(false, a3, false, a3,
                                                  (short)0, c, false, false);
        c = __builtin_amdgcn_wmma_f32_16x16x4_f32(false, a4, false, a4,
                                                  (short)0, c, false, false);
        c = __builtin_amdgcn_wmma_f32_16x16x4_f32(false, a5, false, a5,
                                                  (short)0, c, false, false);
        c = __builtin_amdgcn_wmma_f32_16x16x4_f32(false, a6, false, a6,
                                                  (short)0, c, false, false);
        c = __builtin_amdgcn_wmma_f32_16x16x4_f32(false, a7, false, a7,
                                                  (short)0, c, false, false);
    }

    // C/D layout: vgpr r, lanes 0-15 -> (M=r, N=lane); lanes 16-31 -> (M=r+8, N=lane-16).
    // Chunk-1 diagonal block: lanes 0-7 (M=r<8, N=lane<8).
    // Chunk-2 diagonal block: lanes 24-31 (M=r+8, N=lane-16 in 8..15) -> G[r][lane-24].
    if (lane < 8) {
        #pragma unroll
        for (int r = 0; r < 8; ++r)
            atomicAdd(&lds_g[r * 8 + lane], c[r]);
    } else if (lane >= 24) {
        #pragma unroll
        for (int r = 0; r < 8; ++r)
            atomicAdd(&lds_g[r * 8 + (lane - 24)], c[r]);
    }
    __syncthreads();

    if (tid < 64)
        partials[(size_t)blockIdx.x * 64 + tid] = lds_g[tid];
}

// ---------------------------------------------------------------------------
// Kernel 2: reduce per-WG partial Gram matrices, then run the Frank-Wolfe
// min-norm iteration (tiny 8x8 problem) serially; deterministic.
// ---------------------------------------------------------------------------
__global__ __launch_bounds__(64)
void fw_solver_kernel(const float* __restrict__ partials, int nwg,
                      float* __restrict__ out) {
    __shared__ float G[64];
    const int tid = threadIdx.x;

    float s = 0.0f;
    for (int w = 0; w < nwg; ++w)
        s += partials[(size_t)w * 64 + tid];
    G[tid] = s;
    __syncthreads();

    if (tid == 0) {
        float sol[FW_NTASKS];
        #pragma unroll
        for (int i = 0; i < FW_NTASKS; ++i) sol[i] = 1.0f / FW_NTASKS;

        float change = 3.402823466e+38f;
        for (int it = 0; it < FW_MAXITER && change >= FW_STOP; ++it) {
            float gds[FW_NTASKS];
            #pragma unroll
            for (int i = 0; i < FW_NTASKS; ++i) {
                float acc = 0.0f;
                #pragma unroll
                for (int j = 0; j < FW_NTASKS; ++j)
                    acc += G[i * 8 + j] * sol[j];
                gds[i] = acc;
            }
            // first-occurrence argmin
            int t = 0; float best = gds[0];
            #pragma unroll
            for (int j = 1; j < FW_NTASKS; ++j)
                if (gds[j] < best) { best = gds[j]; t = j; }

            float v1v1 = 0.0f, v1v2 = 0.0f;
            #pragma unroll
            for (int i = 0; i < FW_NTASKS; ++i) {
                v1v1 += sol[i] * gds[i];
                v1v2 += sol[i] * G[i * 8 + t];
            }
            const float v2v2 = G[t * 8 + t];

            float gamma = (v2v2 - v1v2) / (v1v1 + v2v2 - 2.0f * v1v2);
            if (v1v2 >= v2v2) gamma = 0.001f;   // same override order as reference
            if (v1v2 >= v1v1) gamma = 0.999f;

            change = 0.0f;
            #pragma unroll
            for (int i = 0; i < FW_NTASKS; ++i) {
                float ns = gamma * sol[i] + ((i == t) ? (1.0f - gamma) : 0.0f);
                change += fabsf(ns - sol[i]);
                sol[i] = ns;
            }
        }
        #pragma unroll
        for (int i = 0; i < FW_NTASKS; ++i) out[i] = sol[i];
    }
}

extern "C" void kernel_launch(void* const* d_in, const int* in_sizes, int n_in,
                              void* d_out, int out_size, void* d_ws, size_t ws_size,
                              hipStream_t stream) {
    (void)n_in; (void)out_size;
    const float* vecs = (const float*)d_in[0];
    const long D = (long)in_sizes[0] / FW_NTASKS;   // 16777216

    int nwg = 512;
    const size_t need = (size_t)nwg * 64 * sizeof(float);
    if (ws_size < need) {
        nwg = (int)(ws_size / (64 * sizeof(float)));
        if (nwg < 1) nwg = 1;
    }
    float* partials = (float*)d_ws;

    gram_wmma_kernel<<<nwg, 256, 0, stream>>>(vecs, partials, D, nwg);
    fw_solver_kernel<<<1, 64, 0, stream>>>(partials, nwg, (float*)d_out);
}